// GlassBreakSNN_21122649161976
// MI455X (gfx1250) — compile-verified
//
#include <hip/hip_runtime.h>

#ifndef __has_builtin
#define __has_builtin(x) 0
#endif

#if __has_builtin(__builtin_amdgcn_global_load_async_to_lds_b32)
#define USE_ASYNC_LDS 1
#endif

typedef __attribute__((address_space(1))) int gint_t;  // global int
typedef __attribute__((address_space(3))) int lint_t;  // LDS int

__device__ __forceinline__ void async_drain() {
#if defined(USE_ASYNC_LDS)
#if __has_builtin(__builtin_amdgcn_s_wait_asynccnt)
  __builtin_amdgcn_s_wait_asynccnt(0);
#else
  asm volatile("s_wait_asynccnt 0" ::: "memory");
#endif
  asm volatile("" ::: "memory");  // compiler barrier: LDS was modified by async engine
#endif
}

constexpr int T_DIM = 4096;  // timesteps (fixed by reference)
constexpr int ROWS  = 32;    // batch rows per block == one wave32
constexpr int TCH   = 32;    // time chunk (128B coalesced segments)
constexpr int LPAD  = 33;    // padded LDS row stride -> bank-conflict-free both ways

__global__ __launch_bounds__(ROWS) void snn_lif_kernel(
    const float* __restrict__ x,
    const float* __restrict__ w_n1,  const float* __restrict__ w_n2,
    const float* __restrict__ w_n31, const float* __restrict__ w_n32,
    const float* __restrict__ w_inh, const float* __restrict__ w_inh3,
    const float* __restrict__ th1p,  const float* __restrict__ th2p,
    const float* __restrict__ th3p,  const float* __restrict__ thip,
    float* __restrict__ out, int B) {
  const int lane = threadIdx.x;          // 0..31, lane == local batch row
  const int b0   = blockIdx.x * ROWS;

  // Scalar weights/thresholds (uniform -> s_load)
  const float w1  = w_n1[0],  w2  = w_n2[0];
  const float w31 = w_n31[0], w32 = w_n32[0];
  const float wi  = w_inh[0], wi3 = w_inh3[0];
  const float th1 = th1p[0],  th2 = th2p[0];
  const float th3 = th3p[0],  thi = thip[0];
  constexpr float BETA = 0.9f;

  __shared__ float in_t[2][ROWS * LPAD];  // double-buffered input tile
  __shared__ float o1t[ROWS * LPAD];      // output staging tiles
  __shared__ float o2t[ROWS * LPAD];
  __shared__ float o3t[ROWS * LPAD];
  __shared__ float oit[ROWS * LPAD];

  const float* xbase = x + (size_t)b0 * T_DIM;

  // Output layout: [trigger (B)] [s1 (B*T)] [s2] [s3] [si]
  float*       trig = out;
  float*       s1o  = out + B;
  const size_t BT   = (size_t)B * T_DIM;
  float*       s2o  = s1o + BT;
  float*       s3o  = s1o + 2 * BT;
  float*       sio  = s1o + 3 * BT;
  const size_t obase = (size_t)b0 * T_DIM;

  float m1 = 0.f, m2 = 0.f, m3 = 0.f, mi = 0.f, acc3 = 0.f;

  // Fetch one 32x32 input tile: per r, lanes grab 32 consecutive floats (128B)
  auto issue_chunk = [&](int c, int buf) {
    const float* g = xbase + c * TCH + lane;
    float*       l = &in_t[buf][lane];
#if defined(USE_ASYNC_LDS)
#pragma unroll
    for (int r = 0; r < ROWS; ++r) {
      __builtin_amdgcn_global_load_async_to_lds_b32(
          (gint_t*)(g + (size_t)r * T_DIM),
          (lint_t*)(l + r * LPAD),
          /*offset=*/0, /*cpol=*/0);
    }
#else
#pragma unroll
    for (int r = 0; r < ROWS; ++r) l[r * LPAD] = g[(size_t)r * T_DIM];
#endif
  };

  issue_chunk(0, 0);
  async_drain();

  constexpr int NCH = T_DIM / TCH;  // 128 chunks
  for (int c = 0; c < NCH; ++c) {
    // Prefetch next chunk; overlaps with compute + writeout below.
    if (c + 1 < NCH) issue_chunk(c + 1, (c + 1) & 1);

    const float* cin = &in_t[c & 1][lane * LPAD];
    float* p1 = &o1t[lane * LPAD];
    float* p2 = &o2t[lane * LPAD];
    float* p3 = &o3t[lane * LPAD];
    float* pi = &oit[lane * LPAD];

#pragma unroll 8
    for (int tt = 0; tt < TCH; ++tt) {
      const float xv = cin[tt];
      // n1 / n2 / inh update in parallel (ILP=3 dependent chains)
      float r1 = (m1 > th1) ? th1 : 0.f;
      m1 = __builtin_fmaf(BETA, m1, xv * w1) - r1;
      float s1 = (m1 > th1) ? 1.f : 0.f;

      float r2 = (m2 > th2) ? th2 : 0.f;
      m2 = __builtin_fmaf(BETA, m2, xv * w2) - r2;
      float s2 = (m2 > th2) ? 1.f : 0.f;

      float ri = (mi > thi) ? thi : 0.f;
      mi = __builtin_fmaf(BETA, mi, xv * wi) - ri;
      float si = (mi > thi) ? 1.f : 0.f;

      // n3 driven by spikes
      float cur3 = __builtin_fmaf(s1, w31, __builtin_fmaf(s2, w32, si * wi3));
      float r3 = (m3 > th3) ? th3 : 0.f;
      m3 = __builtin_fmaf(BETA, m3, cur3) - r3;
      float s3 = (m3 > th3) ? 1.f : 0.f;

      acc3 += s3;
      p1[tt] = s1; p2[tt] = s2; p3[tt] = s3; pi[tt] = si;
    }

    // Coalesced write-out via LDS transpose (single wave -> DS in-order, no barrier).
    const size_t oc = obase + (size_t)c * TCH + lane;
#pragma unroll
    for (int r = 0; r < ROWS; ++r) {
      const int    li = r * LPAD + lane;
      const size_t gi = oc + (size_t)r * T_DIM;
      __builtin_nontemporal_store(o1t[li], s1o + gi);
      __builtin_nontemporal_store(o2t[li], s2o + gi);
      __builtin_nontemporal_store(o3t[li], s3o + gi);
      __builtin_nontemporal_store(oit[li], sio + gi);
    }

    // Ensure the prefetched tile has landed before next iteration consumes it.
    async_drain();
  }

  // trigger = mean over T of s3
  trig[b0 + lane] = acc3 * (1.0f / (float)T_DIM);
}

extern "C" void kernel_launch(void* const* d_in, const int* in_sizes, int n_in,
                              void* d_out, int out_size, void* d_ws, size_t ws_size,
                              hipStream_t stream) {
  (void)n_in; (void)out_size; (void)d_ws; (void)ws_size;
  const float* x = (const float*)d_in[0];
  const int B = in_sizes[0] / T_DIM;  // C==1 collapses; (B,1,T) -> (B,T)

  dim3 grid(B / ROWS), block(ROWS);
  snn_lif_kernel<<<grid, block, 0, stream>>>(
      x,
      (const float*)d_in[1],  (const float*)d_in[2],
      (const float*)d_in[3],  (const float*)d_in[4],
      (const float*)d_in[5],  (const float*)d_in[6],
      (const float*)d_in[7],  (const float*)d_in[8],
      (const float*)d_in[9],  (const float*)d_in[10],
      (float*)d_out, B);
}